// PostfixNetwork_27393301414038
// MI455X (gfx1250) — compile-verified
//
#include <hip/hip_runtime.h>

#define B_   64
#define S_   512
#define D_   4096
#define K_   8
#define CH   256     // COND_H == SIG_H
#define SF   128     // SIG_F
#define NOUT 32768   // K_*D_

typedef float v2f __attribute__((ext_vector_type(2)));
typedef float v4f __attribute__((ext_vector_type(4)));
typedef float v8f __attribute__((ext_vector_type(8)));

// ---------------------------------------------------------------------------
// Kernel 1: masked mean pool over content slots: pooled[b,d]
// grid (D/256, B), block 256. Coalesced 1KB row segments per step.
// ---------------------------------------------------------------------------
__global__ void pool_kernel(const float* __restrict__ emb,
                            const long long* __restrict__ seqlens,
                            float* __restrict__ pooled) {
    int d = blockIdx.x * blockDim.x + threadIdx.x;
    int b = blockIdx.y;
    int sl = (int)seqlens[b];
    const float* base = emb + (size_t)b * S_ * D_ + d;
    float acc = 0.0f;
    for (int s = 0; s < sl; ++s)
        acc += base[(size_t)s * D_];
    float denom = (float)(sl > 0 ? sl : 1);
    pooled[b * D_ + d] = acc / denom;
}

// ---------------------------------------------------------------------------
// Kernel 2: h = GELU_exact(pooled @ W1 + b1)   [64,4096]x[4096,256]
// grid (4,16) waves of 32; each wave = one 16x16 tile; K loop via
// V_WMMA_F32_16X16X4_F32 (full fp32 precision).
// ---------------------------------------------------------------------------
__global__ void gemm1_gelu_kernel(const float* __restrict__ A,   // pooled [64,4096]
                                  const float* __restrict__ W1,  // [4096,256]
                                  const float* __restrict__ b1,  // [256]
                                  float* __restrict__ h) {       // [64,256]
    int lane = threadIdx.x & 31;
    int mt = blockIdx.x;           // 0..3  (M tiles)
    int nt = blockIdx.y;           // 0..15 (N tiles)
    int lo  = lane & 15;
    int hi  = lane >> 4;
    int kk  = hi * 2;              // this lane's K sub-offset (0 or 2)
    int col = nt * 16 + lo;

    const float* Ap = A  + (size_t)(mt * 16 + lo) * D_ + kk; // A[M=lo][k0+kk..]
    const float* Bp = W1 + (size_t)kk * CH + col;            // B[k0+kk][n=lo]

    v8f c = {};
    for (int k0 = 0; k0 < D_; k0 += 4) {
        v2f a, b;
        a.x = Ap[k0];
        a.y = Ap[k0 + 1];
        b.x = Bp[(size_t)k0 * CH];
        b.y = Bp[(size_t)(k0 + 1) * CH];
        c = __builtin_amdgcn_wmma_f32_16x16x4_f32(false, a, false, b,
                                                  (short)0, c, false, false);
    }
    float bias = b1[col];
#pragma unroll
    for (int v = 0; v < 8; ++v) {
        int row = mt * 16 + v + hi * 8;
        float x = c[v] + bias;
        // exact GELU: 0.5*x*(1+erf(x/sqrt(2)))
        float g = 0.5f * x * (1.0f + erff(x * 0.70710678118654752f));
        h[row * CH + col] = g;
    }
}

// ---------------------------------------------------------------------------
// Kernel 3: sh = SiLU(sigma_features(t) @ sW1 + sb1)   [64,128]x[128,256]
// One block per batch element; features built once in LDS.
// ---------------------------------------------------------------------------
__global__ void sigma_mlp1_kernel(const float* __restrict__ timesteps,
                                  const float* __restrict__ sW1, // [128,256]
                                  const float* __restrict__ sb1, // [256]
                                  float* __restrict__ sh) {      // [64,256]
    __shared__ float feat[SF];
    int b = blockIdx.x;
    int n = threadIdx.x;           // 0..255
    float t = timesteps[b];
    if (n < SF) {
        const int half = SF / 2;   // 64
        int i = n & (half - 1);
        float freq = expf(-logf(10000.0f) * (float)i / (float)half);
        float ang = t * freq;
        feat[n] = (n < half) ? cosf(ang) : sinf(ang);
    }
    __syncthreads();
    float acc = sb1[n];
#pragma unroll 8
    for (int j = 0; j < SF; ++j)
        acc += feat[j] * sW1[j * CH + n];
    sh[b * CH + n] = acc / (1.0f + expf(-acc));   // SiLU
}

// ---------------------------------------------------------------------------
// Kernel 4: postfix = h@W2 + sh@sW2 + b2 + sb2   [64,256]x[256,32768] (x2)
// One wave per 16-col N-tile, owning all 4 M-tiles (B tile loaded once,
// reused across M). Both GEMMs accumulate into the same v8f accumulators.
// ---------------------------------------------------------------------------
__global__ void gemm2_fused_kernel(const float* __restrict__ h,    // [64,256]
                                   const float* __restrict__ W2,   // [256,32768]
                                   const float* __restrict__ b2,   // [32768]
                                   const float* __restrict__ sh,   // [64,256]
                                   const float* __restrict__ sW2,  // [256,32768]
                                   const float* __restrict__ sb2,  // [32768]
                                   float* __restrict__ postfix) {  // [64,32768]
    int lane = threadIdx.x & 31;
    int nt = blockIdx.x;           // 0..2047
    int lo  = lane & 15;
    int hi  = lane >> 4;
    int kk  = hi * 2;
    int col = nt * 16 + lo;

    v8f acc[4] = {};

    // pass 1: h @ W2
    {
        const float* Bp = W2 + (size_t)kk * NOUT + col;
        const float* Ap = h  + (size_t)lo * CH + kk;
        for (int k0 = 0; k0 < CH; k0 += 4) {
            v2f bv;
            bv.x = Bp[(size_t)k0 * NOUT];
            bv.y = Bp[(size_t)(k0 + 1) * NOUT];
#pragma unroll
            for (int mt = 0; mt < 4; ++mt) {
                const float* ap = Ap + (size_t)(mt * 16) * CH + k0;
                v2f av; av.x = ap[0]; av.y = ap[1];
                acc[mt] = __builtin_amdgcn_wmma_f32_16x16x4_f32(
                    false, av, false, bv, (short)0, acc[mt], false, false);
            }
        }
    }
    // pass 2: sh @ sW2 (same accumulators)
    {
        const float* Bp = sW2 + (size_t)kk * NOUT + col;
        const float* Ap = sh  + (size_t)lo * CH + kk;
        for (int k0 = 0; k0 < CH; k0 += 4) {
            v2f bv;
            bv.x = Bp[(size_t)k0 * NOUT];
            bv.y = Bp[(size_t)(k0 + 1) * NOUT];
#pragma unroll
            for (int mt = 0; mt < 4; ++mt) {
                const float* ap = Ap + (size_t)(mt * 16) * CH + k0;
                v2f av; av.x = ap[0]; av.y = ap[1];
                acc[mt] = __builtin_amdgcn_wmma_f32_16x16x4_f32(
                    false, av, false, bv, (short)0, acc[mt], false, false);
            }
        }
    }

    float bias = b2[col] + sb2[col];
#pragma unroll
    for (int mt = 0; mt < 4; ++mt) {
#pragma unroll
        for (int v = 0; v < 8; ++v) {
            int row = mt * 16 + v + hi * 8;
            postfix[(size_t)row * NOUT + col] = acc[mt][v] + bias;
        }
    }
}

// ---------------------------------------------------------------------------
// Kernel 5: out = emb with rows [seqlen, seqlen+K) replaced by postfix.
// The bandwidth-dominant pass (~1 GiB). Nontemporal (TH=NT) float4 streaming
// so the 512 MiB tensor doesn't thrash L2.
// grid (S, B), block 256; one block copies one 16 KB row.
// ---------------------------------------------------------------------------
__global__ void splice_copy_kernel(const float* __restrict__ emb,
                                   const long long* __restrict__ seqlens,
                                   const float* __restrict__ postfix,
                                   float* __restrict__ out) {
    int s = blockIdx.x;
    int b = blockIdx.y;
    int t = threadIdx.x;
    int sl = (int)seqlens[b];

    const v4f* src;
    if (s >= sl && s < sl + K_) {
        src = (const v4f*)(postfix + ((size_t)b * K_ + (s - sl)) * D_);
    } else {
        src = (const v4f*)(emb + ((size_t)b * S_ + s) * D_);
    }
    v4f* dst = (v4f*)(out + ((size_t)b * S_ + s) * D_);

#pragma unroll
    for (int i = 0; i < 4; ++i) {
        int idx = t + i * 256;                        // 1024 float4 per row
        v4f v = __builtin_nontemporal_load(&src[idx]);
        __builtin_nontemporal_store(v, &dst[idx]);
    }
}

// ---------------------------------------------------------------------------
extern "C" void kernel_launch(void* const* d_in, const int* in_sizes, int n_in,
                              void* d_out, int out_size, void* d_ws, size_t ws_size,
                              hipStream_t stream) {
    (void)in_sizes; (void)n_in; (void)out_size; (void)ws_size;

    const float*     emb      = (const float*)d_in[0];
    const long long* seqlens  = (const long long*)d_in[1];  // int64 in reference
    const float*     ts       = (const float*)d_in[2];
    const float*     W1       = (const float*)d_in[3];
    const float*     b1       = (const float*)d_in[4];
    const float*     W2       = (const float*)d_in[5];
    const float*     b2       = (const float*)d_in[6];
    const float*     sW1      = (const float*)d_in[7];
    const float*     sb1      = (const float*)d_in[8];
    const float*     sW2      = (const float*)d_in[9];
    const float*     sb2      = (const float*)d_in[10];
    float*           out      = (float*)d_out;

    char* ws = (char*)d_ws;
    float* pooled  = (float*)ws;                                     // 64*4096 f32 = 1 MiB
    float* h       = (float*)(ws + (size_t)B_ * D_ * 4);             // 64*256  f32
    float* shbuf   = (float*)(ws + (size_t)B_ * D_ * 4 + B_ * CH * 4);
    float* postfix = (float*)(ws + (size_t)B_ * D_ * 4 + 2 * B_ * CH * 4); // 64*32768 f32 = 8 MiB

    pool_kernel<<<dim3(D_ / 256, B_), 256, 0, stream>>>(emb, seqlens, pooled);
    gemm1_gelu_kernel<<<dim3(4, 16), 32, 0, stream>>>(pooled, W1, b1, h);
    sigma_mlp1_kernel<<<B_, CH, 0, stream>>>(ts, sW1, sb1, shbuf);
    gemm2_fused_kernel<<<NOUT / 16, 32, 0, stream>>>(h, W2, b2, shbuf, sW2, sb2, postfix);
    splice_copy_kernel<<<dim3(S_, B_), 256, 0, stream>>>(emb, seqlens, postfix, out);
}